// Model_73778948210773
// MI455X (gfx1250) — compile-verified
//
#include <hip/hip_runtime.h>
#include <math.h>

// ---------------- model dims ----------------
#define E_   128
#define H_   256
#define V_   50000
#define TX_  400
#define TY_  100
#define B_   16
#define EXT_ 100
#define CTX_ 512          // 2H
#define HC_  896          // H + 2H + E
#define G3_  768          // 3H
#define LDC_ 50100        // V + EXT
#define NEG_ (-1000000000.0f)

typedef __attribute__((ext_vector_type(2))) float        v2f;
typedef __attribute__((ext_vector_type(8))) float        v8f;
typedef __attribute__((ext_vector_type(4))) unsigned int v4u;
typedef __attribute__((ext_vector_type(8))) int          v8i;
typedef __attribute__((ext_vector_type(4))) int          v4i;

#define WMMA_F32(a, b, c) \
  __builtin_amdgcn_wmma_f32_16x16x4_f32(false, (a), false, (b), (short)0, (c), false, false)

__device__ __forceinline__ float sigm(float x) { return 1.0f / (1.0f + __expf(-x)); }

// ---------------- embedding gathers ----------------
__global__ void gather_rows(const float* __restrict__ emb, const int* __restrict__ ids,
                            float* __restrict__ out, int nrows) {
  int idx = blockIdx.x * blockDim.x + threadIdx.x;
  if (idx >= nrows * E_) return;
  int r = idx / E_, e = idx % E_;
  out[idx] = emb[(size_t)ids[r] * E_ + e];
}

__global__ void gather_rows_shift(const float* __restrict__ emb, const int* __restrict__ ids,
                                  float* __restrict__ out, int nrows) {
  int idx = blockIdx.x * blockDim.x + threadIdx.x;
  if (idx >= nrows * E_) return;
  int r = idx / E_, e = idx % E_;
  out[idx] = (r < B_) ? 0.0f : emb[(size_t)ids[r - B_] * E_ + e];
}

// -------- TDM: stage a 16 x K fp32 tile (row stride = lda elements) into LDS --------
// D# bitfields per CDNA5 ISA 8.3/8.4: group0 {count, lds_addr, global_addr, type=2},
// group1 {data_size=4B, tensor_dim0=K, tensor_dim1=16, tile_dim0=K, tile_dim1=16,
// dim0_stride=lda}; groups 2/3 zero (2D tensor). 6-arg builtin form (clang-23 lane).
template <int K>
__device__ __forceinline__ void tdm_load_tile(const float* gsrc, int lda, float* sA) {
  unsigned long long ga = (unsigned long long)(const void*)gsrc;
  unsigned int la = (unsigned int)(unsigned long long)(uintptr_t)sA;
  v4u g0;
  g0.x = 1u;                                             // count=1, user descriptor
  g0.y = la;                                             // lds_addr
  g0.z = (unsigned int)(ga & 0xFFFFFFFFull);             // global_addr[31:0]
  g0.w = (unsigned int)((ga >> 32) & 0x01FFFFFFull) | (2u << 30);  // addr[56:32], type=2
  v8i g1;
  g1[0] = (int)0x00020000u;                              // data_size = 4B
  g1[1] = (int)(((unsigned)K & 0xFFFFu) << 16);          // tensor_dim0[15:0]
  g1[2] = (int)(((unsigned)K >> 16) | (16u << 16));      // tensor_dim0[31:16] | tensor_dim1[15:0]
  g1[3] = (int)((unsigned)K << 16);                      // tensor_dim1[31:16]=0 | tile_dim0=K
  g1[4] = 16;                                            // tile_dim1=16, tile_dim2=0
  g1[5] = lda;                                           // tensor_dim0_stride[31:0]
  g1[6] = 0;                                             // stride[47:32] | dim1_stride lo
  g1[7] = 0;
  v4i z4 = {0, 0, 0, 0};
  v8i z8 = {0, 0, 0, 0, 0, 0, 0, 0};
  __builtin_amdgcn_tensor_load_to_lds(g0, g1, z4, z4, z8, 0);
  __builtin_amdgcn_s_wait_tensorcnt(0);
}

// ---------------- generic fp32 WMMA GEMM: C = act(A[M,K] @ W[K,N] + bias) ----
// 4 waves per block, each wave one 16x16 tile; A tile staged once in LDS via TDM.
// grid = (ceil(N/64), M/16), block = 128
template <int K>
__global__ __launch_bounds__(128)
void gemm_f32_wmma(const float* __restrict__ A, int lda,
                   const float* __restrict__ W, int ldw,
                   const float* __restrict__ bias,
                   float* __restrict__ C, int ldc,
                   int N, int act) {
  __shared__ float sA[16 * K];
  if (threadIdx.x < 32)
    tdm_load_tile<K>(A + (size_t)blockIdx.y * 16 * lda, lda, sA);
  __syncthreads();

  int w = threadIdx.x >> 5, L = threadIdx.x & 31;
  int lane16 = L & 15, hi = L >> 4, kb = hi * 2;
  int ncolbase = blockIdx.x * 64 + w * 16;
  if (ncolbase >= N) return;                // wave-uniform guard (EXEC stays all-1s)
  int ncol = ncolbase + lane16;

  v8f c = {};
  for (int k = 0; k < K; k += 4) {
    v2f a = *(const v2f*)&sA[lane16 * K + k + kb];
    v2f b;
    b.x = W[(size_t)(k + kb) * ldw + ncol];
    b.y = W[(size_t)(k + kb + 1) * ldw + ncol];
    c = WMMA_F32(a, b, c);
  }
  float bv = bias ? bias[ncol] : 0.0f;
  for (int i = 0; i < 8; ++i) {
    int row = blockIdx.y * 16 + i + 8 * hi;
    float v = c[i] + bv;
    if (act == 1) v = tanhf(v);
    C[(size_t)row * ldc + ncol] = v;
  }
}

// ---------------- persistent bidirectional-encoder GRU scan -----------------
__global__ __launch_bounds__(512)
void encoder_scan(const float* __restrict__ gi, const float* __restrict__ Wh,
                  const float* __restrict__ bh, const float* __restrict__ maskx,
                  float* __restrict__ enc, float* __restrict__ hfinal, int reverse) {
  __shared__ float hl[B_ * H_];
  int tid = threadIdx.x;
  for (int i = tid; i < B_ * H_; i += 512) hl[i] = 0.0f;
  __syncthreads();
  int w = tid >> 5, L = tid & 31;
  int lane16 = L & 15, hi = L >> 4, kb = hi * 2;
  int col = w * 16 + lane16;

  for (int s = 0; s < TX_; ++s) {
    int m = reverse ? (TX_ - 1 - s) : s;
    v8f cr = {}, cz = {}, cn = {};
    for (int k = 0; k < H_; k += 4) {
      v2f a = *(const v2f*)&hl[lane16 * H_ + k + kb];
      const float* w0 = Wh + (size_t)(k + kb) * G3_;
      const float* w1 = Wh + (size_t)(k + kb + 1) * G3_;
      v2f br, bz, bn;
      br.x = w0[col];           br.y = w1[col];
      bz.x = w0[col + H_];      bz.y = w1[col + H_];
      bn.x = w0[col + 2 * H_];  bn.y = w1[col + 2 * H_];
      cr = WMMA_F32(a, br, cr);
      cz = WMMA_F32(a, bz, cz);
      cn = WMMA_F32(a, bn, cn);
    }
    float hsel8[8];
    const float* gir = gi + (size_t)m * B_ * G3_;
    for (int i = 0; i < 8; ++i) {
      int mb = i + 8 * hi;  // batch row
      float hr = cr[i] + bh[col];
      float hz = cz[i] + bh[col + H_];
      float hn = cn[i] + bh[col + 2 * H_];
      float ir = gir[mb * G3_ + col];
      float iz = gir[mb * G3_ + col + H_];
      float in = gir[mb * G3_ + col + 2 * H_];
      float r = sigm(ir + hr);
      float z = sigm(iz + hz);
      float n = tanhf(in + r * hn);
      float hold = hl[mb * H_ + col];
      float hnew = (1.0f - z) * n + z * hold;
      float mt = maskx[m * B_ + mb];
      float hs = (mt > 0.0f) ? hnew : hold;
      enc[((size_t)m * B_ + mb) * CTX_ + (reverse ? H_ : 0) + col] = (mt > 0.0f) ? hnew : 0.0f;
      hsel8[i] = hs;
    }
    __syncthreads();
    for (int i = 0; i < 8; ++i) hl[(i + 8 * hi) * H_ + col] = hsel8[i];
    __syncthreads();
  }
  for (int i = tid; i < B_ * H_; i += 512) hfinal[i] = hl[i];
}

// ---------------- one decoder GRU cell step (masked blend) ------------------
__global__ __launch_bounds__(512)
void gru_step(const float* __restrict__ x, int Kx, const float* __restrict__ hprev,
              const float* __restrict__ Wi, const float* __restrict__ bi,
              const float* __restrict__ Wh, const float* __restrict__ bh,
              const float* __restrict__ ymrow, float* __restrict__ hout) {
  int tid = threadIdx.x;
  int w = tid >> 5, L = tid & 31;
  int lane16 = L & 15, hi = L >> 4, kb = hi * 2;
  int col = w * 16 + lane16;

  v8f ir8 = {}, iz8 = {}, in8 = {};
  for (int k = 0; k < Kx; k += 4) {
    v2f a;
    a.x = x[(size_t)lane16 * Kx + k + kb];
    a.y = x[(size_t)lane16 * Kx + k + kb + 1];
    const float* w0 = Wi + (size_t)(k + kb) * G3_;
    const float* w1 = Wi + (size_t)(k + kb + 1) * G3_;
    v2f br, bz, bn;
    br.x = w0[col];           br.y = w1[col];
    bz.x = w0[col + H_];      bz.y = w1[col + H_];
    bn.x = w0[col + 2 * H_];  bn.y = w1[col + 2 * H_];
    ir8 = WMMA_F32(a, br, ir8);
    iz8 = WMMA_F32(a, bz, iz8);
    in8 = WMMA_F32(a, bn, in8);
  }
  v8f hr8 = {}, hz8 = {}, hn8 = {};
  for (int k = 0; k < H_; k += 4) {
    v2f a;
    a.x = hprev[(size_t)lane16 * H_ + k + kb];
    a.y = hprev[(size_t)lane16 * H_ + k + kb + 1];
    const float* w0 = Wh + (size_t)(k + kb) * G3_;
    const float* w1 = Wh + (size_t)(k + kb + 1) * G3_;
    v2f br, bz, bn;
    br.x = w0[col];           br.y = w1[col];
    bz.x = w0[col + H_];      bz.y = w1[col + H_];
    bn.x = w0[col + 2 * H_];  bn.y = w1[col + 2 * H_];
    hr8 = WMMA_F32(a, br, hr8);
    hz8 = WMMA_F32(a, bz, hz8);
    hn8 = WMMA_F32(a, bn, hn8);
  }
  for (int i = 0; i < 8; ++i) {
    int mb = i + 8 * hi;
    float ir = ir8[i] + bi[col];
    float iz = iz8[i] + bi[col + H_];
    float in = in8[i] + bi[col + 2 * H_];
    float hr = hr8[i] + bh[col];
    float hz = hz8[i] + bh[col + H_];
    float hn = hn8[i] + bh[col + 2 * H_];
    float r = sigm(ir + hr);
    float z = sigm(iz + hz);
    float n = tanhf(in + r * hn);
    float hold = hprev[mb * H_ + col];
    float hnew = (1.0f - z) * n + z * hold;
    float ym = ymrow[mb];
    hout[mb * H_ + col] = ym * hnew + (1.0f - ym) * hold;
  }
}

// ---------------- misc small kernels ----------------------------------------
__global__ void concat_h(const float* __restrict__ hf, const float* __restrict__ hb,
                         float* __restrict__ out) {
  int idx = blockIdx.x * blockDim.x + threadIdx.x;
  if (idx >= B_ * CTX_) return;
  int b = idx / CTX_, c = idx % CTX_;
  out[idx] = (c < H_) ? hf[b * H_ + c] : hb[b * H_ + c - H_];
}

__global__ void fill_zero(float* p, int n) {
  int idx = blockIdx.x * blockDim.x + threadIdx.x;
  if (idx < n) p[idx] = 0.0f;
}

__global__ __launch_bounds__(128)
void attn_score(const float* __restrict__ whenc, const float* __restrict__ sWs,
                const float* __restrict__ cov, const float* __restrict__ att_Wc,
                const float* __restrict__ att_b, const float* __restrict__ att_V,
                const float* __restrict__ maskx, float* __restrict__ score) {
  __shared__ float red[128];
  int tx = blockIdx.x / B_, b = blockIdx.x % B_;
  int tid = threadIdx.x;
  float covv = cov[b * TX_ + tx];
  const float* wp = whenc + ((size_t)tx * B_ + b) * CTX_;
  float acc = 0.0f;
  for (int c = tid; c < CTX_; c += 128) {
    float st = wp[c] + sWs[b * CTX_ + c] + covv * att_Wc[c] + att_b[c];
    acc += tanhf(st) * att_V[c];
  }
  red[tid] = acc;
  __syncthreads();
  for (int s = 64; s > 0; s >>= 1) {
    if (tid < s) red[tid] += red[tid + s];
    __syncthreads();
  }
  if (tid == 0) {
    float m = maskx[tx * B_ + b];
    score[tx * B_ + b] = (m == 0.0f) ? NEG_ : m * red[0];
  }
}

__global__ __launch_bounds__(256)
void attn_softmax(const float* __restrict__ score, float* __restrict__ cov,
                  float* __restrict__ att_t, float* __restrict__ covs_t) {
  __shared__ float red[256];
  int b = blockIdx.x, tid = threadIdx.x;
  float m = -3.0e38f;
  for (int tx = tid; tx < TX_; tx += 256) m = fmaxf(m, score[tx * B_ + b]);
  red[tid] = m; __syncthreads();
  for (int s = 128; s > 0; s >>= 1) { if (tid < s) red[tid] = fmaxf(red[tid], red[tid + s]); __syncthreads(); }
  float mx = red[0]; __syncthreads();
  float sum = 0.0f;
  for (int tx = tid; tx < TX_; tx += 256) sum += __expf(score[tx * B_ + b] - mx);
  red[tid] = sum; __syncthreads();
  for (int s = 128; s > 0; s >>= 1) { if (tid < s) red[tid] += red[tid + s]; __syncthreads(); }
  float tot = red[0];
  for (int tx = tid; tx < TX_; tx += 256) {
    float wv = __expf(score[tx * B_ + b] - mx) / tot;
    att_t[b * TX_ + tx] = wv;
    covs_t[b * TX_ + tx] = cov[b * TX_ + tx];
    cov[b * TX_ + tx] += wv;
  }
}

__global__ __launch_bounds__(256)
void attn_ctx(const float* __restrict__ att_t, const float* __restrict__ enc,
              float* __restrict__ ctx_t) {
  int b = blockIdx.x, tid = threadIdx.x;
  for (int c = tid; c < CTX_; c += 256) {
    float acc = 0.0f;
    for (int tx = 0; tx < TX_; ++tx)
      acc += att_t[b * TX_ + tx] * enc[((size_t)tx * B_ + b) * CTX_ + c];
    ctx_t[b * CTX_ + c] = acc;
  }
}

__global__ void build_hcat(const float* __restrict__ dec_h, const float* __restrict__ ctxs,
                           const float* __restrict__ ysh, float* __restrict__ hcat) {
  int idx = blockIdx.x * blockDim.x + threadIdx.x;
  if (idx >= TY_ * B_ * HC_) return;
  int row = idx / HC_, c = idx % HC_;
  float v;
  if (c < H_)             v = dec_h[(size_t)row * H_ + c];
  else if (c < H_ + CTX_) v = ctxs[(size_t)row * CTX_ + (c - H_)];
  else                    v = ysh[(size_t)row * E_ + (c - H_ - CTX_)];
  hcat[idx] = v;
}

__global__ __launch_bounds__(256)
void row_softmax(float* __restrict__ C) {
  __shared__ float red[256];
  int row = blockIdx.x, tid = threadIdx.x;
  float* p = C + (size_t)row * LDC_;
  float m = -3.0e38f;
  for (int j = tid; j < V_; j += 256) m = fmaxf(m, p[j]);
  red[tid] = m; __syncthreads();
  for (int s = 128; s > 0; s >>= 1) { if (tid < s) red[tid] = fmaxf(red[tid], red[tid + s]); __syncthreads(); }
  float mx = red[0]; __syncthreads();
  float sum = 0.0f;
  for (int j = tid; j < V_; j += 256) sum += __expf(p[j] - mx);
  red[tid] = sum; __syncthreads();
  for (int s = 128; s > 0; s >>= 1) { if (tid < s) red[tid] += red[tid + s]; __syncthreads(); }
  float tot = red[0];
  for (int j = tid; j < V_; j += 256) p[j] = __expf(p[j] - mx) / tot;
  for (int j = V_ + tid; j < LDC_; j += 256) p[j] = 0.0f;
}

__global__ void g_kernel(const float* __restrict__ hcat, const float* __restrict__ Ww,
                         const float* __restrict__ Wb, float* __restrict__ g) {
  int row = blockIdx.x * blockDim.x + threadIdx.x;
  if (row >= TY_ * B_) return;
  const float* h = hcat + (size_t)row * HC_;
  float acc = Wb[0];
  for (int j = 0; j < HC_; ++j) acc += h[j] * Ww[j];
  g[row] = sigm(acc);
}

__global__ void scale_rows(float* __restrict__ C, const float* __restrict__ g) {
  int idx = blockIdx.x * blockDim.x + threadIdx.x;
  if (idx >= TY_ * B_ * LDC_) return;
  int row = idx / LDC_;
  C[idx] *= g[row];
}

__global__ void scatter_ptr(float* __restrict__ C, const float* __restrict__ g,
                            const float* __restrict__ att, const int* __restrict__ x_ext) {
  int idx = blockIdx.x * blockDim.x + threadIdx.x;
  if (idx >= TY_ * B_ * TX_) return;
  int b = (idx / TX_) % B_, tx = idx % TX_;
  int row = idx / TX_;  // t*B + b
  float val = (1.0f - g[row]) * att[idx];
  int col = x_ext[tx * B_ + b];
  atomicAdd(&C[(size_t)row * LDC_ + col], val);
}

__global__ void cost_kernel(const float* __restrict__ C, const int* __restrict__ y,
                            float* __restrict__ acc) {
  int row = blockIdx.x * blockDim.x + threadIdx.x;
  if (row >= TY_ * B_) return;
  int yv = y[row];
  if (yv != 0) {
    float lp = logf(C[(size_t)row * LDC_ + yv] + 1e-12f);
    atomicAdd(&acc[0], -lp);
    atomicAdd(&acc[1], 1.0f);
  }
}

__global__ __launch_bounds__(256)
void costc_kernel(const float* __restrict__ att, const float* __restrict__ covs,
                  float* __restrict__ acc) {
  __shared__ float red[256];
  int row = blockIdx.x, tid = threadIdx.x;
  float s = 0.0f;
  for (int tx = tid; tx < TX_; tx += 256) {
    size_t i = (size_t)row * TX_ + tx;
    s += fminf(att[i], covs[i]);
  }
  red[tid] = s; __syncthreads();
  for (int st = 128; st > 0; st >>= 1) { if (tid < st) red[tid] += red[tid + st]; __syncthreads(); }
  if (tid == 0) atomicAdd(&acc[2], red[0]);
}

__global__ void final_kernel(const float* __restrict__ acc, float* __restrict__ out2) {
  out2[0] = acc[0] / fmaxf(acc[1], 1.0f);
  out2[1] = acc[2] / (float)(TY_ * B_);
}

// ---------------- host orchestration ----------------------------------------
extern "C" void kernel_launch(void* const* d_in, const int* in_sizes, int n_in,
                              void* d_out, int out_size, void* d_ws, size_t ws_size,
                              hipStream_t stream) {
  const int*   input_x = (const int*)d_in[0];
  const int*   yids    = (const int*)d_in[2];
  const float* mask_x  = (const float*)d_in[3];
  const float* mask_y  = (const float*)d_in[4];
  const int*   x_ext   = (const int*)d_in[5];
  const float* emb     = (const float*)d_in[8];
  const float* enc_f_Wi = (const float*)d_in[9];
  const float* enc_f_Wh = (const float*)d_in[10];
  const float* enc_f_bi = (const float*)d_in[11];
  const float* enc_f_bh = (const float*)d_in[12];
  const float* enc_b_Wi = (const float*)d_in[13];
  const float* enc_b_Wh = (const float*)d_in[14];
  const float* enc_b_bi = (const float*)d_in[15];
  const float* enc_b_bh = (const float*)d_in[16];
  const float* adapt_W  = (const float*)d_in[17];
  const float* adapt_b  = (const float*)d_in[18];
  const float* att_Wh   = (const float*)d_in[19];
  const float* att_Ws   = (const float*)d_in[20];
  const float* att_Wc   = (const float*)d_in[21];
  const float* att_b    = (const float*)d_in[22];
  const float* att_V    = (const float*)d_in[23];
  const float* gru1_Wi  = (const float*)d_in[24];
  const float* gru1_Wh  = (const float*)d_in[25];
  const float* gru1_bi  = (const float*)d_in[26];
  const float* gru1_bh  = (const float*)d_in[27];
  const float* gru2_Wi  = (const float*)d_in[28];
  const float* gru2_Wh  = (const float*)d_in[29];
  const float* gru2_bi  = (const float*)d_in[30];
  const float* gru2_bh  = (const float*)d_in[31];
  const float* V1_W = (const float*)d_in[32];
  const float* V1_b = (const float*)d_in[33];
  const float* V2_W = (const float*)d_in[34];
  const float* V2_b = (const float*)d_in[35];
  const float* W_W  = (const float*)d_in[36];
  const float* W_b  = (const float*)d_in[37];

  float* ws = (float*)d_ws;
  size_t off = 0;
  float* ex       = ws + off; off += (size_t)TX_ * B_ * E_;
  float* ysh      = ws + off; off += (size_t)TY_ * B_ * E_;
  float* gi_f     = ws + off; off += (size_t)TX_ * B_ * G3_;
  float* gi_b     = ws + off; off += (size_t)TX_ * B_ * G3_;
  float* enc      = ws + off; off += (size_t)TX_ * B_ * CTX_;
  float* whenc    = ws + off; off += (size_t)TX_ * B_ * CTX_;
  float* hf       = ws + off; off += (size_t)B_ * H_;
  float* hb       = ws + off; off += (size_t)B_ * H_;
  float* hfb      = ws + off; off += (size_t)B_ * CTX_;
  float* dec_init = ws + off; off += (size_t)B_ * H_;
  float* s1       = ws + off; off += (size_t)B_ * H_;
  float* sWs      = ws + off; off += (size_t)B_ * CTX_;
  float* score    = ws + off; off += (size_t)TX_ * B_;
  float* cov      = ws + off; off += (size_t)B_ * TX_;
  float* att_dist = ws + off; off += (size_t)TY_ * B_ * TX_;
  float* covs     = ws + off; off += (size_t)TY_ * B_ * TX_;
  float* ctxs     = ws + off; off += (size_t)TY_ * B_ * CTX_;
  float* dec_h    = ws + off; off += (size_t)TY_ * B_ * H_;
  float* hcat     = ws + off; off += (size_t)TY_ * B_ * HC_;
  float* t1       = ws + off; off += (size_t)TY_ * B_ * H_;
  float* gbuf     = ws + off; off += (size_t)TY_ * B_;
  float* acc      = ws + off; off += 8;

  float* ypred = (float*)d_out;
  float* costs = ypred + (size_t)TY_ * B_ * LDC_;   // 80,160,000

  // embeddings
  gather_rows<<<(TX_ * B_ * E_ + 255) / 256, 256, 0, stream>>>(emb, input_x, ex, TX_ * B_);
  gather_rows_shift<<<(TY_ * B_ * E_ + 255) / 256, 256, 0, stream>>>(emb, yids, ysh, TY_ * B_);

  // encoder input-gate projections
  gemm_f32_wmma<E_><<<dim3((G3_ + 63) / 64, (TX_ * B_) / 16), 128, 0, stream>>>(
      ex, E_, enc_f_Wi, G3_, enc_f_bi, gi_f, G3_, G3_, 0);
  gemm_f32_wmma<E_><<<dim3((G3_ + 63) / 64, (TX_ * B_) / 16), 128, 0, stream>>>(
      ex, E_, enc_b_Wi, G3_, enc_b_bi, gi_b, G3_, G3_, 0);

  // persistent bi-directional GRU scans
  encoder_scan<<<1, 512, 0, stream>>>(gi_f, enc_f_Wh, enc_f_bh, mask_x, enc, hf, 0);
  encoder_scan<<<1, 512, 0, stream>>>(gi_b, enc_b_Wh, enc_b_bh, mask_x, enc, hb, 1);

  // dec_init = tanh([hf|hb] @ adapt_W + b)
  concat_h<<<(B_ * CTX_ + 255) / 256, 256, 0, stream>>>(hf, hb, hfb);
  gemm_f32_wmma<CTX_><<<dim3((H_ + 63) / 64, 1), 128, 0, stream>>>(
      hfb, CTX_, adapt_W, H_, adapt_b, dec_init, H_, H_, 1);

  // loop-invariant attention keys
  gemm_f32_wmma<CTX_><<<dim3((CTX_ + 63) / 64, (TX_ * B_) / 16), 128, 0, stream>>>(
      enc, CTX_, att_Wh, CTX_, nullptr, whenc, CTX_, CTX_, 0);

  fill_zero<<<(B_ * TX_ + 255) / 256, 256, 0, stream>>>(cov, B_ * TX_);
  fill_zero<<<1, 8, 0, stream>>>(acc, 8);

  // decoder scan
  for (int t = 0; t < TY_; ++t) {
    const float* hidden = (t == 0) ? dec_init : (dec_h + (size_t)(t - 1) * B_ * H_);
    gru_step<<<1, 512, 0, stream>>>(ysh + (size_t)t * B_ * E_, E_, hidden,
                                    gru1_Wi, gru1_bi, gru1_Wh, gru1_bh,
                                    mask_y + t * B_, s1);
    gemm_f32_wmma<H_><<<dim3((CTX_ + 63) / 64, 1), 128, 0, stream>>>(
        s1, H_, att_Ws, CTX_, nullptr, sWs, CTX_, CTX_, 0);
    attn_score<<<TX_ * B_, 128, 0, stream>>>(whenc, sWs, cov, att_Wc, att_b, att_V, mask_x, score);
    attn_softmax<<<B_, 256, 0, stream>>>(score, cov, att_dist + (size_t)t * B_ * TX_,
                                         covs + (size_t)t * B_ * TX_);
    attn_ctx<<<B_, 256, 0, stream>>>(att_dist + (size_t)t * B_ * TX_, enc,
                                     ctxs + (size_t)t * B_ * CTX_);
    gru_step<<<1, 512, 0, stream>>>(ctxs + (size_t)t * B_ * CTX_, CTX_, s1,
                                    gru2_Wi, gru2_bi, gru2_Wh, gru2_bh,
                                    mask_y + t * B_, dec_h + (size_t)t * B_ * H_);
  }

  // pointer-generator head
  build_hcat<<<((TY_ * B_ * HC_) + 255) / 256, 256, 0, stream>>>(dec_h, ctxs, ysh, hcat);
  gemm_f32_wmma<HC_><<<dim3((H_ + 63) / 64, (TY_ * B_) / 16), 128, 0, stream>>>(
      hcat, HC_, V1_W, H_, V1_b, t1, H_, H_, 1);
  gemm_f32_wmma<H_><<<dim3((V_ + 63) / 64, (TY_ * B_) / 16), 128, 0, stream>>>(
      t1, H_, V2_W, V_, V2_b, ypred, LDC_, V_, 0);
  row_softmax<<<TY_ * B_, 256, 0, stream>>>(ypred);
  g_kernel<<<(TY_ * B_ + 255) / 256, 256, 0, stream>>>(hcat, W_W, W_b, gbuf);
  scale_rows<<<((TY_ * B_ * LDC_) + 255) / 256, 256, 0, stream>>>(ypred, gbuf);
  scatter_ptr<<<(TY_ * B_ * TX_ + 255) / 256, 256, 0, stream>>>(ypred, gbuf, att_dist, x_ext);

  // losses
  cost_kernel<<<(TY_ * B_ + 255) / 256, 256, 0, stream>>>(ypred, yids, acc);
  costc_kernel<<<TY_ * B_, 256, 0, stream>>>(att_dist, covs, acc);
  final_kernel<<<1, 1, 0, stream>>>(acc, costs);
}